// MoLoRa_3109556322597
// MI455X (gfx1250) — compile-verified
//
#include <hip/hip_runtime.h>

// MoLoRa fused kernel for gfx1250 (CDNA5, wave32), f32 path via V_WMMA_F32_16X16X4_F32.
// Shapes: B=4 S=2048 D=2048 E=8 R=4 -> N=8192 tokens. ALPHA/R = 4.0.

typedef __attribute__((ext_vector_type(2))) float v2f;
typedef __attribute__((ext_vector_type(8))) float v8f;

#define D_DIM   2048
#define E_NUM   8
#define R_RANK  4
#define ER      32           // E*R
#define N_TOK   8192         // B*S
#define BT_TOK  64           // tokens per workgroup
#define BK      64           // K-chunk for phase A
#define XS_STR  66           // LDS stride (64 + 2 pad) for x chunk rows
#define WS_STR  66           // LDS stride for W^T chunk rows
#define CO_STR  34           // LDS stride (32 + 2 pad) for coeff rows
#define BTT_STR 34           // LDS stride for lora_B^T tile rows
#define OUT_SCALE 4.0f       // ALPHA / R

static __device__ inline v8f zero8() {
  v8f z;
#pragma unroll
  for (int i = 0; i < 8; ++i) z[i] = 0.0f;
  return z;
}

__global__ __launch_bounds__(128) void molora_fused_kernel(
    const float* __restrict__ x,        // [N, D]
    const float* __restrict__ lora_A,   // [E, D, R]
    const float* __restrict__ lora_B,   // [E, R, D] == Bflat [ER, D]
    const float* __restrict__ router_w, // [D, E]
    const float* __restrict__ router_b, // [E]
    float* __restrict__ out)            // [N, D]
{
  __shared__ float xs  [BT_TOK * XS_STR];   // x chunk    [64 rows][66]
  __shared__ float wst [48     * WS_STR];   // W^T chunk  [48 cols][66] (32 A + 8 router + 8 pad)
  __shared__ float coeff[BT_TOK * CO_STR];  // ax -> coeff [64][34]
  __shared__ float lgts[BT_TOK * E_NUM];    // logits     [64][8]
  __shared__ float btsT[64 * BTT_STR];      // lora_B^T tile [64 cols][34]

  const int tid   = threadIdx.x;            // 0..127
  const int lane  = tid & 31;
  const int wave  = tid >> 5;               // 0..3, owns token rows wave*16..+15
  const int lm    = lane & 15;              // row/col-in-tile selector
  const int khalf = lane >> 4;              // selects K pair {0,1} vs {2,3}
  const int tokenBase = blockIdx.x * BT_TOK;

  // ---------------- Phase A: [64 x D] @ [D x 48] -> ax(32) | logits(8) | pad(8)
  v8f acc0 = zero8();  // cols  0..15 (ax)
  v8f acc1 = zero8();  // cols 16..31 (ax)
  v8f acc2 = zero8();  // cols 32..39 logits, 40..47 pad

  for (int kb = 0; kb < D_DIM; kb += BK) {
    __syncthreads();
    // stage x[tokenBase..+63][kb..+63] -> xs (coalesced float4 loads)
#pragma unroll
    for (int i = 0; i < 8; ++i) {
      int idx = tid + i * 128;              // 0..1023
      int row = idx >> 4;
      int c4  = (idx & 15) << 2;
      const float4 v = *(const float4*)(x + (size_t)(tokenBase + row) * D_DIM + kb + c4);
      float* p = &xs[row * XS_STR + c4];
      p[0] = v.x; p[1] = v.y; p[2] = v.z; p[3] = v.w;
    }
    // stage lora_A columns (e*4+r) transposed: wst[c][k]; A[e][kb+k][0..3] is a contiguous float4
#pragma unroll
    for (int i = 0; i < 4; ++i) {
      int idx = tid + i * 128;              // 0..511
      int e   = idx >> 6;
      int kl  = idx & 63;
      const float4 v = *(const float4*)(lora_A + (size_t)e * (D_DIM * R_RANK)
                                        + (size_t)(kb + kl) * R_RANK);
      wst[(e * 4 + 0) * WS_STR + kl] = v.x;
      wst[(e * 4 + 1) * WS_STR + kl] = v.y;
      wst[(e * 4 + 2) * WS_STR + kl] = v.z;
      wst[(e * 4 + 3) * WS_STR + kl] = v.w;
    }
    // stage router_w cols 32..39 and zero pad cols 40..47
    {
      int kl   = tid >> 1;
      int half = tid & 1;
      const float4 v = *(const float4*)(router_w + (size_t)(kb + kl) * E_NUM + half * 4);
      int c = 32 + half * 4;
      wst[(c + 0) * WS_STR + kl] = v.x;
      wst[(c + 1) * WS_STR + kl] = v.y;
      wst[(c + 2) * WS_STR + kl] = v.z;
      wst[(c + 3) * WS_STR + kl] = v.w;
      int cp = 40 + half * 4;
      wst[(cp + 0) * WS_STR + kl] = 0.0f;
      wst[(cp + 1) * WS_STR + kl] = 0.0f;
      wst[(cp + 2) * WS_STR + kl] = 0.0f;
      wst[(cp + 3) * WS_STR + kl] = 0.0f;
    }
    __syncthreads();

    const float* xrow = &xs [(wave * 16 + lm) * XS_STR];
    const float* b0r  = &wst[( 0 + lm) * WS_STR];
    const float* b1r  = &wst[(16 + lm) * WS_STR];
    const float* b2r  = &wst[(32 + lm) * WS_STR];
#pragma unroll
    for (int kk = 0; kk < BK / 4; ++kk) {
      int kc = kk * 4 + khalf * 2;          // even -> 8B aligned ds_load_b64
      v2f a  = *(const v2f*)(xrow + kc);
      v2f b0 = *(const v2f*)(b0r + kc);
      v2f b1 = *(const v2f*)(b1r + kc);
      v2f b2 = *(const v2f*)(b2r + kc);
      acc0 = __builtin_amdgcn_wmma_f32_16x16x4_f32(false, a, false, b0, (short)0, acc0, false, false);
      acc1 = __builtin_amdgcn_wmma_f32_16x16x4_f32(false, a, false, b1, (short)0, acc1, false, false);
      acc2 = __builtin_amdgcn_wmma_f32_16x16x4_f32(false, a, false, b2, (short)0, acc2, false, false);
    }
  }

  // ---------------- dump accumulators: coeff = ax, lgts = logits
  {
    int rBase = wave * 16 + khalf * 8;      // C/D layout: vgpr i, lane l -> row (l/16)*8+i, col l%16
#pragma unroll
    for (int i = 0; i < 8; ++i) {
      coeff[(rBase + i) * CO_STR + lm]      = acc0[i];
      coeff[(rBase + i) * CO_STR + 16 + lm] = acc1[i];
    }
    if (lm < E_NUM) {
#pragma unroll
      for (int i = 0; i < 8; ++i) lgts[(rBase + i) * E_NUM + lm] = acc2[i];
    }
  }
  __syncthreads();

  // ---------------- softmax + fold router prob * (ALPHA/R) into coeff
  if (tid < BT_TOK) {
    float lg[E_NUM];
    float mx = -3.0e38f;
#pragma unroll
    for (int e = 0; e < E_NUM; ++e) {
      lg[e] = lgts[tid * E_NUM + e] + router_b[e];
      mx = fmaxf(mx, lg[e]);
    }
    float s = 0.0f;
#pragma unroll
    for (int e = 0; e < E_NUM; ++e) { lg[e] = expf(lg[e] - mx); s += lg[e]; }
    float inv = OUT_SCALE / s;
#pragma unroll
    for (int c = 0; c < ER; ++c) coeff[tid * CO_STR + c] *= lg[c >> 2] * inv;
  }
  __syncthreads();

  // ---------------- Phase B: out[64 x D] = coeff[64 x 32] @ Bflat[32 x D]
  v2f aF[8];
  {
    const float* crow = &coeff[(wave * 16 + lm) * CO_STR];
#pragma unroll
    for (int kk = 0; kk < 8; ++kk) aF[kk] = *(const v2f*)(crow + kk * 4 + khalf * 2);
  }

  for (int cb = 0; cb < D_DIM; cb += 64) {
    __syncthreads();
    // stage Bflat[0..31][cb..+63] transposed -> btsT[c][k] (coalesced float4 loads)
#pragma unroll
    for (int i = 0; i < 4; ++i) {
      int idx = tid + i * 128;              // 0..511
      int k   = idx >> 4;                   // 0..31
      int c4  = (idx & 15) << 2;
      const float4 v = *(const float4*)(lora_B + (size_t)k * D_DIM + cb + c4);
      btsT[(c4 + 0) * BTT_STR + k] = v.x;
      btsT[(c4 + 1) * BTT_STR + k] = v.y;
      btsT[(c4 + 2) * BTT_STR + k] = v.z;
      btsT[(c4 + 3) * BTT_STR + k] = v.w;
    }
    __syncthreads();

#pragma unroll
    for (int sub = 0; sub < 4; ++sub) {
      v8f d = zero8();
      const float* brow = &btsT[(sub * 16 + lm) * BTT_STR];
#pragma unroll
      for (int kk = 0; kk < 8; ++kk) {
        v2f b = *(const v2f*)(brow + kk * 4 + khalf * 2);
        d = __builtin_amdgcn_wmma_f32_16x16x4_f32(false, aF[kk], false, b, (short)0, d, false, false);
      }
      int col = cb + sub * 16 + lm;
      size_t row0 = (size_t)(tokenBase + wave * 16 + khalf * 8);
#pragma unroll
      for (int i = 0; i < 8; ++i) out[(row0 + i) * D_DIM + col] = d[i];
    }
  }
}

extern "C" void kernel_launch(void* const* d_in, const int* in_sizes, int n_in,
                              void* d_out, int out_size, void* d_ws, size_t ws_size,
                              hipStream_t stream) {
  (void)in_sizes; (void)n_in; (void)out_size; (void)d_ws; (void)ws_size;
  const float* x        = (const float*)d_in[0];
  const float* lora_A   = (const float*)d_in[1];
  const float* lora_B   = (const float*)d_in[2];
  const float* router_w = (const float*)d_in[3];
  const float* router_b = (const float*)d_in[4];
  float* out = (float*)d_out;

  dim3 grid(N_TOK / BT_TOK);   // 128 workgroups
  dim3 block(128);             // 4 waves (wave32)
  molora_fused_kernel<<<grid, block, 0, stream>>>(x, lora_A, lora_B, router_w, router_b, out);
}